// AGCnet_15891378995201
// MI455X (gfx1250) — compile-verified
//
#include <hip/hip_runtime.h>
#include <math.h>

typedef float v2f __attribute__((ext_vector_type(2)));
typedef float v8f __attribute__((ext_vector_type(8)));

#define NB 16
#define NC 64
#define HH 256
#define WW 256
#define PH 128
#define PW 128

// monotonic float <-> uint mapping so atomicMin/Max on uint order floats
__device__ __forceinline__ unsigned f2ord(float f) {
    unsigned u = __float_as_uint(f);
    return (u & 0x80000000u) ? ~u : (u | 0x80000000u);
}
__device__ __forceinline__ float ord2f(unsigned u) {
    return __uint_as_float((u & 0x80000000u) ? (u & 0x7fffffffu) : ~u);
}

__global__ void k_init(unsigned* __restrict__ red) {
    if (threadIdx.x == 0) {
        red[0] = 0xFFFFFFFFu;   // min(x) identity (= ord(+max))
        red[1] = 0x00000000u;   // max(x) identity
        red[2] = 0x80000000u;   // min(x1) seeded with ord(0.0): x1 boundary is exactly 0
        red[3] = 0x80000000u;   // max(x1) seeded with ord(0.0)
    }
}

// 2x2 mean pool fused with global min/max reduction of x (reads x exactly once)
__global__ __launch_bounds__(256) void k_pool_minmax(const float* __restrict__ x,
                                                     float* __restrict__ x0,
                                                     unsigned* __restrict__ red) {
    int t  = blockIdx.x * 256 + threadIdx.x;       // [0, NB*NC*PH*PW)
    int pw = t & (PW - 1);
    int ph = (t >> 7) & (PH - 1);
    int bc = t >> 14;
    const float* base = x + ((size_t)bc * HH + 2 * ph) * WW + 2 * pw;
    float2 a = *(const float2*)(base);
    float2 b = *(const float2*)(base + WW);
    x0[t] = (a.x + a.y + b.x + b.y) * 0.25f;
    float lmin = fminf(fminf(a.x, a.y), fminf(b.x, b.y));
    float lmax = fmaxf(fmaxf(a.x, a.y), fmaxf(b.x, b.y));

    __shared__ float smin[256];
    __shared__ float smax[256];
    int tid = threadIdx.x;
    smin[tid] = lmin; smax[tid] = lmax;
    __syncthreads();
    for (int s = 128; s > 0; s >>= 1) {
        if (tid < s) {
            smin[tid] = fminf(smin[tid], smin[tid + s]);
            smax[tid] = fmaxf(smax[tid], smax[tid + s]);
        }
        __syncthreads();
    }
    if (tid == 0) {
        atomicMin(&red[0], f2ord(smin[0]));
        atomicMax(&red[1], f2ord(smax[0]));
    }
}

// Fused conv3x3 (evaluated only at the odd positions the stride-2 1x1 samples)
// + conv1x1, both as fp32 WMMA 16x16x4 GEMMs. One block = one (b, i, 16-j tile);
// 4 waves each own a 16-output-channel tile.
__global__ __launch_bounds__(128) void k_conv(const float* __restrict__ x0,
                                              const float* __restrict__ w1,
                                              const float* __restrict__ w2,
                                              float* __restrict__ x1buf,
                                              unsigned* __restrict__ red) {
    __shared__ float sx0[NC][3][34];   // 64ch x 3 rows x 33 cols (+pad)
    __shared__ float su[NC][16];       // intermediate u[ic][pos]

    const int bid = blockIdx.x;        // NB * 64 * 4
    const int b   = bid >> 8;
    const int rem = bid & 255;
    const int i   = (rem >> 2) + 1;    // 1..64  -> conv row p = 2i-1
    const int jt  = rem & 3;
    const int j0  = jt * 16 + 1;       // 1,17,33,49

    const int tid   = threadIdx.x;
    const int pbase = 2 * i  - 2;      // rows pbase..pbase+2 (row 128 => zero pad)
    const int qbase = 2 * j0 - 2;      // cols qbase..qbase+32 (col 128 => zero pad)

    for (int idx = tid; idx < NC * 3 * 33; idx += 128) {
        int ic = idx / 99;
        int r  = (idx / 33) % 3;
        int c  = idx % 33;
        int p = pbase + r, q = qbase + c;
        float v = 0.0f;
        if (p < PH && q < PW)
            v = x0[(((size_t)b * NC + ic) * PH + p) * PW + q];
        sx0[ic][r][c] = v;
    }
    __syncthreads();

    const int wave = tid >> 5;         // oc tile 0..3
    const int lane = tid & 31;
    const int half = lane >> 4;        // A: lanes 16-31 hold K=2,3 ; B rows 2/vgpr
    const int lm   = lane & 15;
    const int oc   = wave * 16 + lm;   // A-matrix row fed by this lane

    // GEMM1: u[oc][n] = sum_{tap,ic} w1[oc][ic][tap] * patch[ic][tap][2n+dx]
    // K ordered tap-major (tap*64 + ic) so each K=4 chunk stays in one tap.
    v8f acc = {};
    const float* w1row = w1 + (size_t)oc * 576;
    for (int tap = 0; tap < 9; ++tap) {
        const int dy = tap / 3, dx = tap - dy * 3;
        for (int icb = 0; icb < NC; icb += 4) {
            v2f a, bb;
            a.x  = w1row[(icb + 2 * half    ) * 9 + tap];
            a.y  = w1row[(icb + 2 * half + 1) * 9 + tap];
            bb.x = sx0[icb + half    ][dy][2 * lm + dx];
            bb.y = sx0[icb + 2 + half][dy][2 * lm + dx];
            acc = __builtin_amdgcn_wmma_f32_16x16x4_f32(false, a, false, bb,
                                                        (short)0, acc, false, false);
        }
    }
    // D layout: lane holds D[v + half*8][lm] in acc[v]
    #pragma unroll
    for (int v = 0; v < 8; ++v)
        su[wave * 16 + v + half * 8][lm] = acc[v];
    __syncthreads();

    // GEMM2: x1[oc][n] = sum_ic w2[oc][ic] * u[ic][n]   (K = 64)
    v8f acc2 = {};
    const float* w2row = w2 + (size_t)oc * 64;
    for (int icb = 0; icb < NC; icb += 4) {
        v2f a, bb;
        a.x  = w2row[icb + 2 * half];
        a.y  = w2row[icb + 2 * half + 1];
        bb.x = su[icb + half    ][lm];
        bb.y = su[icb + 2 + half][lm];
        acc2 = __builtin_amdgcn_wmma_f32_16x16x4_f32(false, a, false, bb,
                                                     (short)0, acc2, false, false);
    }

    float lmin = 3.4e38f, lmax = -3.4e38f;
    #pragma unroll
    for (int v = 0; v < 8; ++v) {
        int oco = wave * 16 + v + half * 8;
        float val = acc2[v];
        x1buf[(((size_t)b * NC + oco) * 64 + (i - 1)) * 64 + (jt * 16 + lm)] = val;
        lmin = fminf(lmin, val);
        lmax = fmaxf(lmax, val);
    }
    for (int off = 16; off > 0; off >>= 1) {
        lmin = fminf(lmin, __shfl_xor(lmin, off, 32));
        lmax = fmaxf(lmax, __shfl_xor(lmax, off, 32));
    }
    if (lane == 0) {
        atomicMin(&red[2], f2ord(lmin));
        atomicMax(&red[3], f2ord(lmax));
    }
}

// Streaming pass: y = x + (x4*(max-min)+min); x44 = exposure(x4, per-quadrant gamma).
// Quadrants (0,0),(0,1),(1,0) use gamma from x1n[0,0]==x1n[0,1]==x1n[1,0] (all from
// x1==0 at padded positions); only (1,1) uses the real conv value x1[b,c,1,1].
__global__ __launch_bounds__(256) void k_final(const float* __restrict__ x,
                                               const float* __restrict__ x1buf,
                                               const unsigned* __restrict__ red,
                                               float* __restrict__ out) {
    const size_t n = (size_t)NB * NC * HH * WW;
    size_t t = (size_t)blockIdx.x * 256 + threadIdx.x;   // one float4 each
    size_t e = t * 4;
    int w  = (int)(e & (WW - 1));        // 4-wide group never straddles w=128
    int h  = (int)((e >> 8) & (HH - 1));
    int bc = (int)(e >> 16);

    float xmin = ord2f(red[0]);
    float xmax = ord2f(red[1]);
    float m1   = ord2f(red[2]);
    float M1   = ord2f(red[3]);
    float scale = xmax - xmin;
    float inv   = 1.0f / scale;
    float ginv  = 2.0f / (M1 - m1);
    float g0    = (0.0f - m1) * ginv;
    float g11   = (x1buf[(size_t)bc * 4096] - m1) * ginv;   // x1[b,c,1,1]

    float g = ((h >= 128) && (w >= 128)) ? g11 : g0;

    float4 xv = *(const float4*)(x + e);
    float4 yv, av;
    {
        float d;
        d = (xv.x - xmin) * inv; yv.x = xv.x + (d * scale + xmin);
        av.x = (g < 1.0f) ? g * log2f(1.0f + d) : powf(d, g);
        d = (xv.y - xmin) * inv; yv.y = xv.y + (d * scale + xmin);
        av.y = (g < 1.0f) ? g * log2f(1.0f + d) : powf(d, g);
        d = (xv.z - xmin) * inv; yv.z = xv.z + (d * scale + xmin);
        av.z = (g < 1.0f) ? g * log2f(1.0f + d) : powf(d, g);
        d = (xv.w - xmin) * inv; yv.w = xv.w + (d * scale + xmin);
        av.w = (g < 1.0f) ? g * log2f(1.0f + d) : powf(d, g);
    }
    *(float4*)(out + e)     = yv;
    *(float4*)(out + n + e) = av;
}

extern "C" void kernel_launch(void* const* d_in, const int* in_sizes, int n_in,
                              void* d_out, int out_size, void* d_ws, size_t ws_size,
                              hipStream_t stream) {
    const float* x  = (const float*)d_in[0];
    const float* w1 = (const float*)d_in[1];
    const float* w2 = (const float*)d_in[2];
    float* out = (float*)d_out;

    unsigned* red = (unsigned*)d_ws;                       // 4 reduction words
    float* x0     = (float*)d_ws + 16;                     // 16*64*128*128 = 64 MB
    float* x1buf  = x0 + (size_t)NB * NC * PH * PW;        // 16*64*64*64  = 16 MB

    k_init<<<1, 32, 0, stream>>>(red);
    k_pool_minmax<<<(NB * NC * PH * PW) / 256, 256, 0, stream>>>(x, x0, red);
    k_conv<<<NB * 64 * 4, 128, 0, stream>>>(x0, w1, w2, x1buf, red);
    k_final<<<(int)(((size_t)NB * NC * HH * WW / 4) / 256), 256, 0, stream>>>(x, x1buf, red, out);
}